// RoIAlignAvg_82016695484581
// MI455X (gfx1250) — compile-verified
//
#include <hip/hip_runtime.h>

typedef float v2f __attribute__((ext_vector_type(2)));
typedef float v8f __attribute__((ext_vector_type(8)));

#define NC 1024
#define NH 50
#define NW 50

// One workgroup (256 threads = 8 waves) per RoI.
// Separable RoIAlign+AvgPool: pooled = Wx * (Wy * G)^T per channel, where
// G is the 16x16 bilinear-corner gather grid of this RoI. Both 16-deep
// contractions run on V_WMMA_F32_16X16X4_F32 (4 k-chunks each), full f32.
__global__ __launch_bounds__(256) void roialign_pool_wmma(
    const float* __restrict__ feat,   // (B, 1024, 50, 50)
    const float* __restrict__ rois,   // (K, 5)
    float* __restrict__ out,          // (K, 1024, 7, 7)
    int nroi)
{
  __shared__ float CY[16], CX[16];   // corner weights (validity folded; CX has 0.25 pool factor)
  __shared__ int   YC[16], XC[16];   // corner coordinates
  __shared__ float Tsh[8][16][16];   // per-wave stage-1 tile (T[iy][xi]) re-layout buffer

  const int roi  = blockIdx.x;
  const int tid  = threadIdx.x;
  const int lane = tid & 31;
  const int wave = tid >> 5;
  const int m    = lane & 15;   // M/N index within 16 for WMMA operands
  const int hi   = lane >> 4;   // lane half selects K sub-rows {0,1} vs {2,3}

  // ---- per-RoI sampling metadata (threads 0..7: y-axis, 8..15: x-axis) ----
  if (tid < 16) {
    const int  t   = tid & 7;
    const bool isY = tid < 8;
    const float p1 = rois[roi * 5 + (isY ? 2 : 1)] * 0.0625f;
    const float p2 = rois[roi * 5 + (isY ? 4 : 3)] * 0.0625f;
    const float bs = fmaxf(p2 - p1, 0.0f) * (1.0f / 7.0f);
    const float g  = p1 + (float)t * bs;           // sample coordinate t of 8
    const bool  v  = (g >= 0.0f) && (g < 50.0f);   // reference validity
    int i0 = (int)floorf(g);
    i0 = min(max(i0, 0), NH - 2);                  // clip like reference
    const float w  = g - (float)i0;                // may be <0 or >1 (extrapolation), as in ref
    const float w0 = v ? (1.0f - w) : 0.0f;
    const float w1 = v ? w : 0.0f;
    if (isY) {
      CY[2 * t] = w0;          CY[2 * t + 1] = w1;
      YC[2 * t] = i0;          YC[2 * t + 1] = i0 + 1;
    } else {
      CX[2 * t] = 0.25f * w0;  CX[2 * t + 1] = 0.25f * w1;  // fold pooling 0.25
      XC[2 * t] = i0;          XC[2 * t + 1] = i0 + 1;
    }
  }
  __syncthreads();

  const int bidx = (int)rois[roi * 5 + 0];

  // ---- channel-invariant WMMA A operands + gather offsets (kept in VGPRs) ----
  // A layout (16x4 f32): reg0 <-> K = 4*kc + 2*hi, reg1 <-> K = 4*kc + 2*hi + 1, M = lane%16.
  // Wy[iy, yi] nonzero iff yi in [2*iy, 2*iy+3] (2x2 pool over adjacent sample rows).
  v2f ay[4], ax[4];
  int goff[4][2];
  const int xc = XC[m];
  #pragma unroll
  for (int kc = 0; kc < 4; ++kc) {
    #pragma unroll
    for (int r = 0; r < 2; ++r) {
      const int k = 4 * kc + 2 * hi + r;
      const bool nz = (m < 7) && (k >= 2 * m) && (k <= 2 * m + 3);
      ay[kc][r] = nz ? CY[k] : 0.0f;
      ax[kc][r] = nz ? CX[k] : 0.0f;
      goff[kc][r] = YC[k] * NW + xc;   // B gather: G[yi=k][xi=m] (x-coalesced across lanes)
    }
  }

  // ---- channel loop: wave w handles c = w, w+8, ... (128 iters) ----
  for (int c = wave; c < NC; c += 8) {
    const float* fp = feat + ((size_t)bidx * NC + c) * (NH * NW);

    // Stage 1: T(iy, xi) = Wy @ G   (contraction over 16 yi, 4 WMMA k-chunks)
    v8f acc = {0.f, 0.f, 0.f, 0.f, 0.f, 0.f, 0.f, 0.f};
    #pragma unroll
    for (int kc = 0; kc < 4; ++kc) {
      v2f bg;
      bg[0] = fp[goff[kc][0]];
      bg[1] = fp[goff[kc][1]];
      acc = __builtin_amdgcn_wmma_f32_16x16x4_f32(false, ay[kc], false, bg,
                                                  (short)0, acc, false, false);
    }
    if (c + 8 < NC) __builtin_prefetch(fp + 8 * (NH * NW) + goff[0][0]);

    // Re-layout D tile (M=iy in vgpr idx, N=xi in lane) -> B operand via wave-private LDS
    #pragma unroll
    for (int v = 0; v < 8; ++v) Tsh[wave][v + 8 * hi][m] = acc[v];

    // Stage 2: pooled(jx, iy) = Wx @ T^T  (contraction over 16 xi)
    v8f p = {0.f, 0.f, 0.f, 0.f, 0.f, 0.f, 0.f, 0.f};
    #pragma unroll
    for (int kc = 0; kc < 4; ++kc) {
      v2f bt;
      bt[0] = Tsh[wave][m][4 * kc + 2 * hi + 0];   // B: (K=xi, N=iy=m) = T[iy][xi]
      bt[1] = Tsh[wave][m][4 * kc + 2 * hi + 1];
      p = __builtin_amdgcn_wmma_f32_16x16x4_f32(false, ax[kc], false, bt,
                                                (short)0, p, false, false);
    }

    // D2(M=jx, N=iy): lanes 0..6 hold iy=lane; vgpr v holds jx=v. 49 floats contiguous.
    float* op = out + (((size_t)roi * NC + c) * 7 + lane) * 7;
    if (lane < 7) {
      #pragma unroll
      for (int j = 0; j < 7; ++j) op[j] = p[j];
    }
  }
}

extern "C" void kernel_launch(void* const* d_in, const int* in_sizes, int n_in,
                              void* d_out, int out_size, void* d_ws, size_t ws_size,
                              hipStream_t stream) {
  const float* feat = (const float*)d_in[0];
  const float* rois = (const float*)d_in[1];
  float* out = (float*)d_out;
  const int nroi = in_sizes[1] / 5;   // K = 512
  roialign_pool_wmma<<<dim3(nroi), dim3(256), 0, stream>>>(feat, rois, out, nroi);
}